// WeightedRigidAlign_927712936253
// MI455X (gfx1250) — compile-verified
//
#include <hip/hip_runtime.h>

typedef float v2f __attribute__((ext_vector_type(2)));
typedef float v8f __attribute__((ext_vector_type(8)));

#define NBATCH   64
#define NTHREADS 256
#define PPT      4                 // points per thread per chunk
#define CHUNK    (NTHREADS * PPT)  // 1024 points per block-chunk
#define LDS_STRIDE 17              // pad 16 -> 17 to kill bank conflicts

// ---------------------------------------------------------------------------
// Kernel 0: zero the 16-float-per-batch accumulator region in d_ws
// ---------------------------------------------------------------------------
__global__ __launch_bounds__(NTHREADS) void wra_zero(float* __restrict__ acc) {
    int t = threadIdx.x;
#pragma unroll
    for (int i = 0; i < 4; ++i) acc[t + i * NTHREADS] = 0.0f;
}

// ---------------------------------------------------------------------------
// Kernel 1: streaming weighted reduction.
// Per batch b, accumulate 16 values:
//   s[0]       = sum(w)
//   s[1..3]    = sum(w * tc)
//   s[4..6]    = sum(w * pc)
//   s[7+3i+j]  = sum((w * tc_i) * pc_j)
// Block tail: 256x16 partials -> LDS -> wave0 column-sums via
// v_wmma_f32_16x16x4_f32 with all-ones B -> 16 global f32 atomics.
// ---------------------------------------------------------------------------
__global__ __launch_bounds__(NTHREADS) void wra_reduce(
    const float* __restrict__ pred, const float* __restrict__ truec,
    const float* __restrict__ wgt, const unsigned char* __restrict__ mask,
    float* __restrict__ acc, int M, int chunks)
{
    const int b = blockIdx.y;
    const float* __restrict__ pb = pred  + (size_t)b * M * 3;
    const float* __restrict__ tb = truec + (size_t)b * M * 3;
    const float* __restrict__ wb = wgt   + (size_t)b * M;
    const unsigned char* __restrict__ mb = mask + (size_t)b * M;

    float s[16];
#pragma unroll
    for (int i = 0; i < 16; ++i) s[i] = 0.0f;

    for (int chunk = blockIdx.x; chunk < chunks; chunk += gridDim.x) {
        const int m0 = (chunk * NTHREADS + threadIdx.x) * PPT;
        // 4 points = 12 floats per coord array; m0 % 4 == 0 so 3*m0 % 4 == 0
        const float4* p4 = (const float4*)(pb + (size_t)m0 * 3);
        const float4* t4 = (const float4*)(tb + (size_t)m0 * 3);
        float4 P0 = p4[0], P1 = p4[1], P2 = p4[2];
        float4 T0 = t4[0], T1 = t4[1], T2 = t4[2];
        float4 W4 = *(const float4*)(wb + m0);
        unsigned int mm = *(const unsigned int*)(mb + m0);

        float pf[12] = {P0.x, P0.y, P0.z, P0.w, P1.x, P1.y, P1.z, P1.w, P2.x, P2.y, P2.z, P2.w};
        float tf[12] = {T0.x, T0.y, T0.z, T0.w, T1.x, T1.y, T1.z, T1.w, T2.x, T2.y, T2.z, T2.w};
        float wf[4]  = {W4.x, W4.y, W4.z, W4.w};

#pragma unroll
        for (int i = 0; i < PPT; ++i) {
            const float wt = ((mm >> (8 * i)) & 0xffu) ? wf[i] : 0.0f;
            const float tx = tf[3 * i + 0], ty = tf[3 * i + 1], tz = tf[3 * i + 2];
            const float px = pf[3 * i + 0], py = pf[3 * i + 1], pz = pf[3 * i + 2];
            const float a0 = wt * tx, a1 = wt * ty, a2 = wt * tz;
            s[0] += wt;
            s[1] += a0;  s[2] += a1;  s[3] += a2;
            s[4] += wt * px;  s[5] += wt * py;  s[6] += wt * pz;
            s[7]  += a0 * px;  s[8]  += a0 * py;  s[9]  += a0 * pz;
            s[10] += a1 * px;  s[11] += a1 * py;  s[12] += a1 * pz;
            s[13] += a2 * px;  s[14] += a2 * py;  s[15] += a2 * pz;
        }
    }

    // ---- block reduction: 256 rows x 16 cols in LDS, wave0 sums columns ----
    __shared__ float lds[NTHREADS * LDS_STRIDE];
#pragma unroll
    for (int j = 0; j < 16; ++j) lds[threadIdx.x * LDS_STRIDE + j] = s[j];
    __syncthreads();

    if (threadIdx.x < 32) {
        const int lane = threadIdx.x;
        const int col  = lane & 15;   // == A-matrix row M (accumulator index)
        const int half = lane >> 4;   // which pair of K-slots this lane feeds
        v8f c = {};
        v2f bones; bones.x = 1.0f; bones.y = 1.0f;   // B = ones(4x16)
        for (int row = 0; row < NTHREADS; row += 4) {
            v2f a;
            a.x = lds[(row + 2 * half + 0) * LDS_STRIDE + col];
            a.y = lds[(row + 2 * half + 1) * LDS_STRIDE + col];
            // D[m][n] += sum_k A[m][k] : column sums of the partials matrix
            c = __builtin_amdgcn_wmma_f32_16x16x4_f32(
                    false, a, false, bones, (short)0, c, false, false);
        }
        // D layout: VGPR r, lanes 0-15 -> M=r ; lanes 16-31 -> M=8+r
        float* ab = acc + b * 16;
        if (lane == 0) {
#pragma unroll
            for (int r = 0; r < 8; ++r) atomicAdd(&ab[r], c[r]);
        } else if (lane == 16) {
#pragma unroll
            for (int r = 0; r < 8; ++r) atomicAdd(&ab[8 + r], c[r]);
        }
    }
}

// ---------------------------------------------------------------------------
// Kernel 2: per-batch 3x3 Kabsch solve (64 threads, 1 block).
// cov = S_tp - (S_t)(S_p)^T / wsum ; SVD via Jacobi on cov^T cov ;
// rot = V diag(1,1,det(V U^T)) U^T
// ---------------------------------------------------------------------------
__device__ __forceinline__ float det3(const float m[3][3]) {
    return m[0][0] * (m[1][1] * m[2][2] - m[1][2] * m[2][1])
         - m[0][1] * (m[1][0] * m[2][2] - m[1][2] * m[2][0])
         + m[0][2] * (m[1][0] * m[2][1] - m[1][1] * m[2][0]);
}

__global__ __launch_bounds__(64) void wra_solve(const float* __restrict__ acc,
                                                float* __restrict__ sol) {
    const int b = threadIdx.x;
    if (b >= NBATCH) return;
    const float* s = acc + b * 16;

    const float wsum = s[0];
    const float inv  = 1.0f / wsum;
    float tbar[3] = {s[1] * inv, s[2] * inv, s[3] * inv};
    float pbar[3] = {s[4] * inv, s[5] * inv, s[6] * inv};

    float cov[3][3];
#pragma unroll
    for (int i = 0; i < 3; ++i)
#pragma unroll
        for (int j = 0; j < 3; ++j)
            cov[i][j] = s[7 + 3 * i + j] - s[1 + i] * s[4 + j] * inv;

    // A = cov^T cov (symmetric PSD), Jacobi eigendecomposition -> V, lambda
    float A[3][3];
#pragma unroll
    for (int i = 0; i < 3; ++i)
#pragma unroll
        for (int j = 0; j < 3; ++j)
            A[i][j] = cov[0][i] * cov[0][j] + cov[1][i] * cov[1][j] + cov[2][i] * cov[2][j];

    float V[3][3] = {{1, 0, 0}, {0, 1, 0}, {0, 0, 1}};
    for (int sweep = 0; sweep < 10; ++sweep) {
#pragma unroll
        for (int pair = 0; pair < 3; ++pair) {
            const int p = (pair == 2) ? 1 : 0;
            const int q = (pair == 0) ? 1 : 2;
            const float apq = A[p][q];
            if (fabsf(apq) > 1e-24f) {
                const float tau = (A[q][q] - A[p][p]) / (2.0f * apq);
                const float t = (tau >= 0.0f ? 1.0f : -1.0f) /
                                (fabsf(tau) + sqrtf(1.0f + tau * tau));
                const float cs = 1.0f / sqrtf(1.0f + t * t);
                const float sn = t * cs;
#pragma unroll
                for (int k = 0; k < 3; ++k) {   // A <- A J  (columns p,q)
                    const float akp = A[k][p], akq = A[k][q];
                    A[k][p] = cs * akp - sn * akq;
                    A[k][q] = sn * akp + cs * akq;
                }
#pragma unroll
                for (int k = 0; k < 3; ++k) {   // A <- J^T A (rows p,q)
                    const float apk = A[p][k], aqk = A[q][k];
                    A[p][k] = cs * apk - sn * aqk;
                    A[q][k] = sn * apk + cs * aqk;
                }
#pragma unroll
                for (int k = 0; k < 3; ++k) {   // V <- V J
                    const float vkp = V[k][p], vkq = V[k][q];
                    V[k][p] = cs * vkp - sn * vkq;
                    V[k][q] = sn * vkp + cs * vkq;
                }
            }
        }
    }

    float lam[3] = {A[0][0], A[1][1], A[2][2]};
    // sort eigenpairs descending (columns of V follow)
#pragma unroll
    for (int i = 0; i < 2; ++i)
#pragma unroll
        for (int j = 0; j < 2 - i; ++j)
            if (lam[j] < lam[j + 1]) {
                float tl = lam[j]; lam[j] = lam[j + 1]; lam[j + 1] = tl;
#pragma unroll
                for (int k = 0; k < 3; ++k) {
                    float tv = V[k][j]; V[k][j] = V[k][j + 1]; V[k][j + 1] = tv;
                }
            }

    // U columns: u_c = cov * v_c / sigma_c (normalize by actual norm)
    float U[3][3];
#pragma unroll
    for (int c = 0; c < 3; ++c) {
        float u0 = cov[0][0] * V[0][c] + cov[0][1] * V[1][c] + cov[0][2] * V[2][c];
        float u1 = cov[1][0] * V[0][c] + cov[1][1] * V[1][c] + cov[1][2] * V[2][c];
        float u2 = cov[2][0] * V[0][c] + cov[2][1] * V[1][c] + cov[2][2] * V[2][c];
        const float n = sqrtf(u0 * u0 + u1 * u1 + u2 * u2);
        if (c == 2 && n < 1e-12f) {   // rank-deficient fallback: u2 = u0 x u1
            u0 = U[1][0] * U[2][1] - U[2][0] * U[1][1];
            u1 = U[2][0] * U[0][1] - U[0][0] * U[2][1];
            u2 = U[0][0] * U[1][1] - U[1][0] * U[0][1];
            U[0][c] = u0; U[1][c] = u1; U[2][c] = u2;
        } else {
            const float rn = 1.0f / fmaxf(n, 1e-30f);
            U[0][c] = u0 * rn; U[1][c] = u1 * rn; U[2][c] = u2 * rn;
        }
    }

    const float d3 = det3(U) * det3(V);   // det(V U^T)

    float* o = sol + b * 16;
#pragma unroll
    for (int i = 0; i < 3; ++i)
#pragma unroll
        for (int j = 0; j < 3; ++j)
            o[3 * i + j] = V[i][0] * U[j][0] + V[i][1] * U[j][1] + d3 * V[i][2] * U[j][2];
    o[9]  = tbar[0]; o[10] = tbar[1]; o[11] = tbar[2];
    o[12] = pbar[0]; o[13] = pbar[1]; o[14] = pbar[2];
    o[15] = 0.0f;
}

// ---------------------------------------------------------------------------
// Kernel 3: apply  out = rot @ (masked(tc) - tbar) + pbar   (streaming)
// ---------------------------------------------------------------------------
__global__ __launch_bounds__(NTHREADS) void wra_apply(
    const float* __restrict__ truec, const unsigned char* __restrict__ mask,
    const float* __restrict__ sol, float* __restrict__ out, int M)
{
    const int b = blockIdx.y;
    const float* o = sol + b * 16;
    const float r00 = o[0], r01 = o[1], r02 = o[2];
    const float r10 = o[3], r11 = o[4], r12 = o[5];
    const float r20 = o[6], r21 = o[7], r22 = o[8];
    const float t0 = o[9], t1 = o[10], t2 = o[11];
    const float p0 = o[12], p1 = o[13], p2 = o[14];

    const int m0 = (blockIdx.x * NTHREADS + threadIdx.x) * PPT;
    if (m0 >= M) return;

    const float* tb = truec + (size_t)b * M * 3;
    const unsigned char* mb = mask + (size_t)b * M;
    float* ob = out + (size_t)b * M * 3;

    const float4* t4 = (const float4*)(tb + (size_t)m0 * 3);
    float4 T0 = t4[0], T1 = t4[1], T2 = t4[2];
    unsigned int mm = *(const unsigned int*)(mb + m0);

    float tf[12] = {T0.x, T0.y, T0.z, T0.w, T1.x, T1.y, T1.z, T1.w, T2.x, T2.y, T2.z, T2.w};
    float g[12];
#pragma unroll
    for (int i = 0; i < PPT; ++i) {
        float x = tf[3 * i + 0], y = tf[3 * i + 1], z = tf[3 * i + 2];
        if (!((mm >> (8 * i)) & 0xffu)) { x = 0.0f; y = 0.0f; z = 0.0f; }
        const float dx = x - t0, dy = y - t1, dz = z - t2;
        g[3 * i + 0] = fmaf(r00, dx, fmaf(r01, dy, fmaf(r02, dz, p0)));
        g[3 * i + 1] = fmaf(r10, dx, fmaf(r11, dy, fmaf(r12, dz, p1)));
        g[3 * i + 2] = fmaf(r20, dx, fmaf(r21, dy, fmaf(r22, dz, p2)));
    }
    float4* o4 = (float4*)(ob + (size_t)m0 * 3);
    o4[0] = make_float4(g[0], g[1], g[2], g[3]);
    o4[1] = make_float4(g[4], g[5], g[6], g[7]);
    o4[2] = make_float4(g[8], g[9], g[10], g[11]);
}

// ---------------------------------------------------------------------------
extern "C" void kernel_launch(void* const* d_in, const int* in_sizes, int n_in,
                              void* d_out, int out_size, void* d_ws, size_t ws_size,
                              hipStream_t stream) {
    const float* pred  = (const float*)d_in[0];
    const float* truec = (const float*)d_in[1];
    const float* wgt   = (const float*)d_in[2];
    const unsigned char* mask = (const unsigned char*)d_in[3];
    float* out = (float*)d_out;

    const int B = NBATCH;
    const int M = in_sizes[2] / B;          // weights are [B, M]
    const int chunks = M / CHUNK;           // M = 262144 -> 256 chunks

    float* acc = (float*)d_ws;              // 16 floats per batch (1024 f32)
    float* sol = acc + 16 * NBATCH;         // rot(9)+tbar(3)+pbar(3)+pad

    wra_zero<<<1, NTHREADS, 0, stream>>>(acc);

    dim3 g1(64, B);
    wra_reduce<<<g1, NTHREADS, 0, stream>>>(pred, truec, wgt, mask, acc, M, chunks);

    wra_solve<<<1, 64, 0, stream>>>(acc, sol);

    dim3 g3(chunks, B);
    wra_apply<<<g3, NTHREADS, 0, stream>>>(truec, mask, sol, out, M);
}